// GlacierDynamicsCheckpointed_74921409512056
// MI455X (gfx1250) — compile-verified
//
#include <hip/hip_runtime.h>
#include <cstdint>

// ---------------------------------------------------------------------------
// Glacier SIA solver for MI455X (gfx1250), compile-only tuned.
//
// Roofline: ~4 GFLOP total, ~6 MB live state -> fully L2-resident (192 MB);
// latency/launch-bound, FP32 stencil with serial 400-step chain => no WMMA
// structure. CDNA5 paths used: per-lane global_load_async_to_lds_b128
// (ASYNCcnt) halo staging in BOTH stencil kernels + s_wait_asynccnt, plus a
// fused last-block max-reduction (atomic ticket) to drop the per-step launch
// count from 3 to 2 (800 launches total, graph-captured).
// ---------------------------------------------------------------------------

#define NYX 512
#define NN  (NYX * NYX)      // 262144 cells
#define NSG (NYX - 1)        // 511 staggered grid (D stored padded to stride 512)
#define N_STEPS 400
#define N_PRECIP 64

__device__ __constant__ float kDX    = 200.0f;
__device__ __constant__ float kInvDX = 1.0f / 200.0f;

struct State {
    float time, t_last, dt;
    int   idx, s1, s2, cap1, cap2, do_upd;
};

// FD * (RHO*G)^3 evaluated in double then demoted (matches JAX weak typing).
__device__ __forceinline__ float cfd() {
    constexpr double rg = 910.0 * 9.81;
    constexpr double v  = 1e-16 * rg * rg * rg;
    return (float)v;
}

// Low 32 bits of a generic LDS pointer == addrspace(3) byte offset.
__device__ __forceinline__ unsigned lds_off(const void* p) {
    return (unsigned)(unsigned long long)(uintptr_t)p;
}

__device__ __forceinline__ void async_tile_b128(unsigned ldsAddr, const float* gptr) {
    unsigned long long g = (unsigned long long)(uintptr_t)gptr;
    // GV mode: VDST = LDS byte offset, VADDR = 64-bit global address.
    asm volatile("global_load_async_to_lds_b128 %0, %1, off"
                 :: "v"(ldsAddr), "v"(g) : "memory");
}

__device__ __forceinline__ void wait_asynccnt0() {
    asm volatile("s_wait_asynccnt 0" ::: "memory");
}

// ---------------------------------------------------------------------------
// Kernel A: diffusivity D (padded stride 512) + fused global max -> dt/state.
// 16x16 output tile; async-stages 17x20 H and Zs tiles into LDS. Last block
// (atomic ticket) reduces the 1024 block maxima and advances the scalar
// state, then resets the ticket (deterministic across graph replays).
// ---------------------------------------------------------------------------
__global__ void kA(const float* __restrict__ H, const float* __restrict__ Zs,
                   float* __restrict__ D, float* __restrict__ part,
                   State* st, unsigned* ticket) {
    __shared__ __align__(16) float sH[17 * 20];
    __shared__ __align__(16) float sZ[17 * 20];
    __shared__ float red[256];
    __shared__ int  isLast;

    const int tx = threadIdx.x, ty = threadIdx.y;
    const int t  = ty * 16 + tx;
    const int c0 = blockIdx.x * 16, r0 = blockIdx.y * 16;

    if (t < 85) {                      // waves 0..2 stage H: 17 rows x 5 segs
        int row = t / 5, seg = t % 5;
        async_tile_b128(lds_off(&sH[row * 20 + seg * 4]),
                        H + (size_t)(r0 + row) * NYX + c0 + seg * 4);
    } else if (t >= 128 && t < 213) {  // waves 4..6 stage Zs
        int u = t - 128, row = u / 5, seg = u % 5;
        async_tile_b128(lds_off(&sZ[row * 20 + seg * 4]),
                        Zs + (size_t)(r0 + row) * NYX + c0 + seg * 4);
    }
    wait_asynccnt0();
    __syncthreads();

    const int r = r0 + ty, c = c0 + tx;
    float d = 0.0f;
    if (r < NSG && c < NSG) {
        float h00 = sH[ty * 20 + tx],       h01 = sH[ty * 20 + tx + 1];
        float h10 = sH[(ty + 1) * 20 + tx], h11 = sH[(ty + 1) * 20 + tx + 1];
        float z00 = sZ[ty * 20 + tx],       z01 = sZ[ty * 20 + tx + 1];
        float z10 = sZ[(ty + 1) * 20 + tx], z11 = sZ[(ty + 1) * 20 + tx + 1];
        float havg = 0.25f * (h00 + h11 + h01 + h10);
        float sx = 0.5f * ((z01 - z00) + (z11 - z10)) * kInvDX;
        float sy = 0.5f * ((z10 - z00) + (z11 - z01)) * kInvDX;
        float sn2 = sx * sx + sy * sy + 1e-10f;   // Snorm^2 (sqrt then **2)
        float h2 = havg * havg, h4 = h2 * h2;
        d = cfd() * h4 * havg * sn2 + 1e-10f;
        D[r * NYX + c] = d;                        // padded stride for B128 tiles
    }
    red[t] = d;
    __syncthreads();
    for (int s = 128; s > 0; s >>= 1) {
        if (t < s) red[t] = fmaxf(red[t], red[t + s]);
        __syncthreads();
    }
    if (t == 0) {
        part[blockIdx.y * gridDim.x + blockIdx.x] = red[0];
        __threadfence();
        unsigned prev = atomicAdd(ticket, 1u);
        isLast = (prev == 1023u) ? 1 : 0;
    }
    __syncthreads();

    if (isLast) {
        volatile const float* vp = part;           // force past-L0 reads
        float m = fmaxf(fmaxf(vp[t], vp[t + 256]),
                        fmaxf(vp[t + 512], vp[t + 768]));
        red[t] = m;
        __syncthreads();
        for (int s = 128; s > 0; s >>= 1) {
            if (t < s) red[t] = fmaxf(red[t], red[t + s]);
            __syncthreads();
        }
        if (t == 0) {
            float maxD = red[0];
            float time = st->time;
            bool active = time < 200.0f;
            float dt = fminf((kDX * kDX) / (2.7f * maxD), 1.0f);
            if (!active) dt = 0.0f;
            float nt = time + dt;
            int s1 = st->s1, s2 = st->s2;
            int cap1 = (!s1 && nt >= 120.0f) ? 1 : 0;
            int cap2 = (!s2 && nt >= 160.0f) ? 1 : 0;
            int du   = (active && (nt - st->t_last) >= 5.0f) ? 1 : 0;
            st->cap1 = cap1;     st->cap2 = cap2;     st->do_upd = du;
            st->s1 = s1 | cap1;  st->s2 = s2 | cap2;
            if (du) { st->idx = min(st->idx + 1, N_PRECIP - 1); st->t_last = nt; }
            st->time = nt;       st->dt = dt;
            *ticket = 0;                           // rearm for next step
            __threadfence();
        }
    }
}

// ---------------------------------------------------------------------------
// Kernel C: flux divergence + ReLU update. Async-stages the 17x24 D halo and
// 18x24 Zs halo (aligned start at col c0-4). Double-buffered Zs avoids the
// 5-point read/write race. Snapshots go straight into d_out.
// ---------------------------------------------------------------------------
__global__ void kC(float* __restrict__ H,
                   const float* __restrict__ Zc, float* __restrict__ Zn,
                   float* __restrict__ smb, const float* __restrict__ D,
                   const float* __restrict__ Zt, const float* __restrict__ precip,
                   const float* __restrict__ Tm, const float* __restrict__ Ts,
                   const State* __restrict__ st, float* __restrict__ out) {
    __shared__ __align__(16) float sD[17 * 24];
    __shared__ __align__(16) float sZ[18 * 24];

    const int tx = threadIdx.x, ty = threadIdx.y;
    const int t  = ty * 16 + tx;
    const int c0 = blockIdx.x * 16, r0 = blockIdx.y * 16;

    // Halo tiles start at (r0-1, c0-4); out-of-range reads land in adjacent
    // ws regions and are never consumed (interior threads index in-range).
    if (t < 102) {                     // D: 17 rows x 6 segs
        int row = t / 6, seg = t % 6;
        async_tile_b128(lds_off(&sD[row * 24 + seg * 4]),
                        D + (size_t)(r0 - 1 + row) * NYX + (c0 - 4) + seg * 4);
    } else if (t >= 128 && t < 236) {  // Zs: 18 rows x 6 segs
        int u = t - 128, row = u / 6, seg = u % 6;
        async_tile_b128(lds_off(&sZ[row * 24 + seg * 4]),
                        Zc + (size_t)(r0 - 1 + row) * NYX + (c0 - 4) + seg * 4);
    }
    wait_asynccnt0();
    __syncthreads();

    const int c = c0 + tx, r = r0 + ty;
    const int rc = r * NYX + c;
    const float dt = st->dt;

    float h = H[rc];
    if (r >= 1 && r < NYX - 1 && c >= 1 && c < NYX - 1) {
        // D row (r-1) -> tile row ty; col (c-1) -> tile col tx+3.
        float Dtl = sD[ty * 24 + tx + 3],       Dtr = sD[ty * 24 + tx + 4];
        float Dbl = sD[(ty + 1) * 24 + tx + 3], Dbr = sD[(ty + 1) * 24 + tx + 4];
        // Zs row (r-1) -> tile row ty; col (c-1) -> tile col tx+3.
        float zu = sZ[ty * 24 + tx + 4];
        float zl = sZ[(ty + 1) * 24 + tx + 3];
        float zc = sZ[(ty + 1) * 24 + tx + 4];
        float zr = sZ[(ty + 1) * 24 + tx + 5];
        float zd = sZ[(ty + 2) * 24 + tx + 4];
        float qxp = -0.5f * (Dtr + Dbr) * (zr - zc) * kInvDX;
        float qxm = -0.5f * (Dtl + Dbl) * (zc - zl) * kInvDX;
        float qyp = -0.5f * (Dbl + Dbr) * (zd - zc) * kInvDX;
        float qym = -0.5f * (Dtl + Dtr) * (zc - zu) * kInvDX;
        float dHdt = -((qxp - qxm) + (qyp - qym)) * kInvDX;
        h = fmaxf(0.0f, h + dt * (dHdt + smb[rc]));   // squash_smb fwd = id
        H[rc] = h;
    }
    float z = Zt[rc] + h;                              // squash_zs fwd = id
    Zn[rc] = z;
    if (st->cap1) out[rc] = h;
    if (st->cap2) out[NN + rc] = h;
    if (st->do_upd) {
        float T = Tm[0] + Ts[0] - 0.006f * z;
        smb[rc] = precip[(size_t)st->idx * NN + rc] - 0.5f * fmaxf(T, 0.0f);
    }
}

__global__ void kInit(float* __restrict__ H, float* __restrict__ Zs,
                      float* __restrict__ smb, const float* __restrict__ Zt,
                      const float* __restrict__ precip, const float* __restrict__ Tm,
                      const float* __restrict__ Ts, State* st, unsigned* ticket,
                      float* __restrict__ out) {
    const int i = blockIdx.x * blockDim.x + threadIdx.x;
    float zt = Zt[i];
    H[i] = 0.0f;
    Zs[i] = zt;
    float T = Tm[0] + Ts[0] - 0.006f * zt;
    smb[i] = precip[i] - 0.5f * fmaxf(T, 0.0f);
    out[i] = 0.0f;
    out[NN + i] = 0.0f;
    if (i == 0) {
        st->time = 0.0f; st->t_last = 0.0f; st->dt = 0.0f;
        st->idx = 0; st->s1 = 0; st->s2 = 0;
        st->cap1 = 0; st->cap2 = 0; st->do_upd = 0;
        *ticket = 0u;
    }
}

__global__ void kFin(const float* __restrict__ H, float* __restrict__ out) {
    const int i = blockIdx.x * blockDim.x + threadIdx.x;
    out[2 * NN + i] = H[i];
}

// ---------------------------------------------------------------------------
// ws layout (floats): H | ZsA | ZsB | smb | D(stride 512, NN slot) |
//                     part(1024) | State | ticket
// Contiguity matters: halo B128 loads may overshoot into the next region.
// ---------------------------------------------------------------------------
extern "C" void kernel_launch(void* const* d_in, const int* in_sizes, int n_in,
                              void* d_out, int out_size, void* d_ws, size_t ws_size,
                              hipStream_t stream) {
    (void)in_sizes; (void)n_in; (void)out_size; (void)ws_size;
    const float* Zt     = (const float*)d_in[0];
    const float* precip = (const float*)d_in[1];
    const float* Tm     = (const float*)d_in[2];
    const float* Ts     = (const float*)d_in[3];
    float* out = (float*)d_out;
    float* ws  = (float*)d_ws;

    float*    H      = ws;
    float*    Zs0    = ws + (size_t)NN;
    float*    Zs1    = ws + (size_t)2 * NN;
    float*    smb    = ws + (size_t)3 * NN;
    float*    D      = ws + (size_t)4 * NN;
    float*    part   = ws + (size_t)5 * NN;
    State*    st     = (State*)(ws + (size_t)5 * NN + 1024);
    unsigned* ticket = (unsigned*)(ws + (size_t)5 * NN + 1024 + 16);

    kInit<<<NN / 256, 256, 0, stream>>>(H, Zs0, smb, Zt, precip, Tm, Ts, st, ticket, out);

    int cur = 0;
    for (int s = 0; s < N_STEPS; ++s) {
        float* Zc = cur ? Zs1 : Zs0;
        float* Zn = cur ? Zs0 : Zs1;
        kA<<<dim3(32, 32), dim3(16, 16), 0, stream>>>(H, Zc, D, part, st, ticket);
        kC<<<dim3(32, 32), dim3(16, 16), 0, stream>>>(H, Zc, Zn, smb, D, Zt,
                                                      precip, Tm, Ts, st, out);
        cur ^= 1;
    }
    kFin<<<NN / 256, 256, 0, stream>>>(H, out);
}